// FasterRCNN_64931315581598
// MI455X (gfx1250) — compile-verified
//
#include <hip/hip_runtime.h>
#include <hip/hip_bf16.h>
#include <math.h>

#define TPB 256
#define MGT 32
#define IOU_THRESH 0.5f

typedef float v4f __attribute__((ext_vector_type(4)));   // native vector for NT load

// ---- CDNA5 async global->LDS path (guarded; falls back to plain LDS store) ----
#if defined(__HIP_DEVICE_COMPILE__) && __has_builtin(__builtin_amdgcn_global_load_async_to_lds_b32)
#define HAVE_ASYNC_LDS 1
#else
#define HAVE_ASYNC_LDS 0
#endif

#if HAVE_ASYNC_LDS
typedef __attribute__((address_space(1))) int g_int;   // global
typedef __attribute__((address_space(3))) int l_int;   // LDS
#endif

__global__ __launch_bounds__(TPB) void rcnn_match_kernel(
    const float* __restrict__ anchors,  // (B, N, 4)
    const float* __restrict__ gts,      // (B, M=32, 5)
    float* __restrict__ out,            // (B, N, 5)
    int N)
{
    // GT record: {x0, y0, x1, y1, area, cls, pad, pad} -> two b128 LDS loads per GT
    __shared__ __align__(16) float gtp[MGT * 8];
    __shared__ __align__(16) float res[TPB * 5];

    const int tid = threadIdx.x;
    const int b  = blockIdx.y;          // batch on grid-Y: no SALU div in prologue
    const int n0 = blockIdx.x * TPB;
    const int anchor = n0 + tid;

    // Issue the anchor load early so it overlaps GT staging.
    // Streamed once -> non-temporal (TH=NT) so it doesn't churn cache.
    const v4f a = __builtin_nontemporal_load(
        reinterpret_cast<const v4f*>(anchors) + (size_t)b * N + anchor);

    // Prefetch GT row for this batch (emits global_prefetch_b8 on gfx1250).
    __builtin_prefetch(gts + (size_t)b * MGT * 5, 0, 3);

    // ---- Stage 32x5 GT floats into LDS via async-to-LDS DMA path ----
    if (tid < MGT * 5) {
        const int i = tid / 5;            // GT index
        const int j = tid % 5;            // component
        const int dst_slot = i * 8 + (j < 4 ? j : 5);   // cls goes to slot 5
        const float* src = gts + ((size_t)b * MGT + i) * 5 + j;
#if HAVE_ASYNC_LDS
        __builtin_amdgcn_global_load_async_to_lds_b32(
            (g_int*)(uintptr_t)src,
            (l_int*)&gtp[dst_slot],
            0, 0);
#else
        gtp[dst_slot] = *src;
#endif
    }
#if HAVE_ASYNC_LDS
#if __has_builtin(__builtin_amdgcn_s_wait_asynccnt)
    __builtin_amdgcn_s_wait_asynccnt(0);
#else
    asm volatile("s_wait_asynccnt 0" ::: "memory");
#endif
#endif
    __syncthreads();

    // Precompute per-GT areas: area2 = |(x0-x1)*(y0-y1)|
    if (tid < MGT) {
        const float x0 = gtp[tid * 8 + 0];
        const float y0 = gtp[tid * 8 + 1];
        const float x1 = gtp[tid * 8 + 2];
        const float y1 = gtp[tid * 8 + 3];
        gtp[tid * 8 + 4] = fabsf((x0 - x1) * (y0 - y1));
    }
    __syncthreads();

    // ---- Per-anchor scan over 32 GTs (LDS broadcast reads) ----
    const float area1 = fabsf((a.x - a.z) * (a.y - a.w));
    float bestq = -INFINITY;
    int   besti = 0;
    bool  any_valid = false;

#pragma unroll 8
    for (int i = 0; i < MGT; ++i) {
        const v4f g  = *reinterpret_cast<const v4f*>(&gtp[i * 8]);      // ds_load_b128
        const v4f ac = *reinterpret_cast<const v4f*>(&gtp[i * 8 + 4]);  // {area, cls, pad, pad}
        const float area2 = ac.x;
        const bool  valid = (ac.y != -1.0f);
        any_valid |= valid;

        const float xl = fmaxf(a.x, g.x);
        const float yb = fmaxf(a.y, g.y);
        const float xr = fminf(a.z, g.z);
        const float yt = fminf(a.w, g.w);
        const float dx = xr - xl;
        const float dy = yt - yb;
        const float inter = fabsf(dx * dy);
        float iou = inter / (area1 + area2 - inter);
        if (dx < 0.0f || dy < 0.0f) iou = 0.0f;   // no_inter mask
        iou = valid ? iou : -1.0f;                // invalid-GT mask

        if (iou > bestq) { bestq = iou; besti = i; }   // strict > == first-max (jnp.argmax)
    }

    // Threshold logic: IOU_LOW == IOU_HIGH == 0.5 -> neutral band is empty.
    // q <= 0.5 (or no valid GT at all) -> all -1; else gather winning GT (4 coords + cls).
    float m0, m1, m2, m3, m4;
    if (!any_valid || bestq <= IOU_THRESH) {
        m0 = m1 = m2 = m3 = m4 = -1.0f;
    } else {
        m0 = gtp[besti * 8 + 0];
        m1 = gtp[besti * 8 + 1];
        m2 = gtp[besti * 8 + 2];
        m3 = gtp[besti * 8 + 3];
        m4 = gtp[besti * 8 + 5];   // class
    }

    // Stage results in LDS (stride-5 word writes: 5 coprime with 64 banks -> conflict-free)
    res[tid * 5 + 0] = m0;
    res[tid * 5 + 1] = m1;
    res[tid * 5 + 2] = m2;
    res[tid * 5 + 3] = m3;
    res[tid * 5 + 4] = m4;
    __syncthreads();

    // Fully coalesced, non-temporal writeback: block owns a contiguous 1280-float span.
    const size_t outbase = ((size_t)b * N + n0) * 5;
#pragma unroll
    for (int j = 0; j < 5; ++j) {
        __builtin_nontemporal_store(res[j * TPB + tid],
                                    out + outbase + (size_t)j * TPB + tid);
    }
}

extern "C" void kernel_launch(void* const* d_in, const int* in_sizes, int n_in,
                              void* d_out, int out_size, void* d_ws, size_t ws_size,
                              hipStream_t stream) {
    const float* anchors = (const float*)d_in[0];   // (B, N, 4) f32
    const float* gts     = (const float*)d_in[1];   // (B, 32, 5) f32
    float*       out     = (float*)d_out;           // (B, N, 5) f32

    const int B = in_sizes[1] / (MGT * 5);          // 1280 / 160 = 8
    const int N = in_sizes[0] / (B * 4);            // 65536

    dim3 grid(N / TPB, B);                          // (256, 8)
    rcnn_match_kernel<<<grid, TPB, 0, stream>>>(anchors, gts, out, N);
}